// GraphNeuralNetwork_39771397161525
// MI455X (gfx1250) — compile-verified
//
#include <hip/hip_runtime.h>
#include <hip/hip_bf16.h>

typedef __attribute__((ext_vector_type(2))) float v2f;
typedef __attribute__((ext_vector_type(8))) float v8f;

#define N_NODES 50000
#define D 128
#define EDGES 600000
#define LDW 132   // LDS dword stride: (4*m + x) mod 64 distinct across lanes -> conflict-free

// ---------------- Kernel 1: zero the accumulator workspace ----------------
__global__ __launch_bounds__(256) void zero_agg(float4* __restrict__ p, int n4) {
    int i = blockIdx.x * 256 + threadIdx.x;
    if (i < n4) p[i] = make_float4(0.f, 0.f, 0.f, 0.f);
}

// ---------------- Kernel 2: SpMM scatter (one edge per wave32) ----------------
__global__ __launch_bounds__(256) void spmm_scatter(
    const float* __restrict__ feat, const int* __restrict__ rows,
    const int* __restrict__ cols, const float* __restrict__ vals,
    float* __restrict__ agg, int nedges) {
    int wave = threadIdx.x >> 5;
    int lane = threadIdx.x & 31;
    int e = blockIdx.x * 8 + wave;
    if (e >= nedges) return;
    int   r = rows[e];
    int   c = cols[e];
    float v = vals[e];
    // 32 lanes x float4 = full 128-float row, coalesced 512B
    float4 f = ((const float4*)(feat + (size_t)c * D))[lane];
    float* dst = agg + (size_t)r * D + lane * 4;
    unsafeAtomicAdd(dst + 0, v * f.x);
    unsafeAtomicAdd(dst + 1, v * f.y);
    unsafeAtomicAdd(dst + 2, v * f.z);
    unsafeAtomicAdd(dst + 3, v * f.w);
}

// ------- Kernel 3: fp32 WMMA GEMM (16x128 stripe / block) + ReLU + L2 norm -------
__global__ __launch_bounds__(256) void gemm_relu_norm(
    const float* __restrict__ agg, const float* __restrict__ Wg,
    float* __restrict__ out) {
    __shared__ float As[16 * LDW];   // 16x128 A tile (fp32)
    __shared__ float Os[16 * LDW];   // 16x128 relu'd output tile

    const int tid    = threadIdx.x;
    const int m_base = blockIdx.x * 16;

    // Stage A tile: 16 rows x 32 float4 = 512 float4 loads over 256 threads
    for (int i = tid; i < 16 * 32; i += 256) {
        int r  = i >> 5;
        int c4 = i & 31;
        float4 a = ((const float4*)(agg + (size_t)(m_base + r) * D))[c4];
        float* d = &As[r * LDW + c4 * 4];
        d[0] = a.x; d[1] = a.y; d[2] = a.z; d[3] = a.w;
    }
    __syncthreads();

    const int wave   = tid >> 5;
    const int lane   = tid & 31;
    const int half   = lane >> 4;   // selects K pair within the fragment
    const int ml     = lane & 15;   // M (for A) / N (for B) index
    const int n_base = wave * 16;   // each wave owns 16 output columns

    v8f acc = {};
    #pragma unroll
    for (int k = 0; k < D; k += 4) {
        // A 16x4 fp32 fragment: lanes 0-15 -> K={k,k+1}, lanes 16-31 -> K={k+2,k+3}
        v2f a = *(const v2f*)&As[ml * LDW + k + 2 * half];
        // B 4x16 fp32 fragment (W row-major, cached in WGP$/L2), mirrored layout
        v2f b;
        b.x = Wg[(k + 2 * half    ) * D + n_base + ml];
        b.y = Wg[(k + 2 * half + 1) * D + n_base + ml];
        acc = __builtin_amdgcn_wmma_f32_16x16x4_f32(
            /*neg_a=*/false, a, /*neg_b=*/false, b,
            /*c_mod=*/(short)0, acc, /*reuse_a=*/false, /*reuse_b=*/false);
    }

    // ReLU + park the 16x16 tile in LDS (C layout: VGPR r -> M = r + 8*half, N = ml)
    #pragma unroll
    for (int r = 0; r < 8; ++r) {
        float v = acc[r] > 0.f ? acc[r] : 0.f;
        Os[(r + 8 * half) * LDW + n_base + ml] = v;
    }
    __syncthreads();

    // Row-wise L2 normalize: 16 threads per row, 8 elements each
    const int row = tid >> 4;
    const int seg = tid & 15;
    const float* o = &Os[row * LDW + seg * 8];
    float s = 0.f;
    #pragma unroll
    for (int j = 0; j < 8; ++j) s += o[j] * o[j];
    #pragma unroll
    for (int off = 8; off >= 1; off >>= 1)
        s += __shfl_xor(s, off, 16);          // threads sharing a row sit in one 16-lane half
    float inv = 1.f / fmaxf(sqrtf(s), 1e-12f);

    float* og = out + (size_t)(m_base + row) * D + seg * 8;
    #pragma unroll
    for (int j = 0; j < 8; ++j) og[j] = o[j] * inv;
}

extern "C" void kernel_launch(void* const* d_in, const int* in_sizes, int n_in,
                              void* d_out, int out_size, void* d_ws, size_t ws_size,
                              hipStream_t stream) {
    const float* feat = (const float*)d_in[0];
    const int*   rows = (const int*)  d_in[1];
    const int*   cols = (const int*)  d_in[2];
    const float* vals = (const float*)d_in[3];
    const float* W    = (const float*)d_in[4];
    float*       out  = (float*)d_out;
    float*       agg  = (float*)d_ws;            // N_NODES*128 fp32 = 25.6 MB scratch

    const int n4 = N_NODES * D / 4;              // 1.6M float4
    zero_agg<<<(n4 + 255) / 256, 256, 0, stream>>>((float4*)agg, n4);
    spmm_scatter<<<(EDGES + 7) / 8, 256, 0, stream>>>(feat, rows, cols, vals, agg, EDGES);
    gemm_relu_norm<<<N_NODES / 16, 256, 0, stream>>>(agg, W, out);
}